// SharedLITNet_58016418234966
// MI455X (gfx1250) — compile-verified
//
#include <hip/hip_runtime.h>
#include <cstdint>

#define BB   4
#define CIN  3
#define HH   128
#define WW   128
#define PP   16384   // H*W
#define HID  64
#define MM   256
#define KC   32
#define KCP  36      // padded LDS row stride: 16B-aligned, bank-conflict-free

typedef __attribute__((ext_vector_type(2))) float v2f;
typedef __attribute__((ext_vector_type(4))) float v4f;
typedef __attribute__((ext_vector_type(8))) float v8f;

// CDNA5 async global->LDS (ASYNCcnt-tracked), per CDNA5 ISA
#define ASYNC_COPY_B128(g, l)                                                  \
  asm volatile("global_load_async_to_lds_b128 %0, %1, off" ::"v"(              \
                   (uint32_t)(uintptr_t)(l)),                                  \
               "v"(g) : "memory")
#define ASYNC_WAIT(n) asm volatile("s_wait_asynccnt " #n ::: "memory")

__device__ __forceinline__ float gelu_exact(float v) {
  return 0.5f * v * (1.0f + erff(v * 0.70710678118654752f));
}

__device__ __forceinline__ v8f wmma_f32(v2f a, v2f b, v8f c) {
  // D = A(16x4 f32) * B(4x16 f32) + C(16x16 f32)
  return __builtin_amdgcn_wmma_f32_16x16x4_f32(false, a, false, b, (short)0, c,
                                               false, false);
}

// ---------------------------------------------------------------- lift (1x1)
__global__ void k_lift(const float* __restrict__ x, const float* __restrict__ lw,
                       const float* __restrict__ lb, float* __restrict__ h) {
  int idx = blockIdx.x * blockDim.x + threadIdx.x;   // over B*P
  int b = idx >> 14, p = idx & (PP - 1);
  int ph = p >> 7, pw = p & (WW - 1);
  float in5[5];
  in5[0] = x[(b * CIN + 0) * PP + p];
  in5[1] = x[(b * CIN + 1) * PP + p];
  in5[2] = x[(b * CIN + 2) * PP + p];
  in5[3] = (float)ph * (1.0f / 127.0f);
  in5[4] = (float)pw * (1.0f / 127.0f);
#pragma unroll 8
  for (int o = 0; o < HID; ++o) {
    float acc = lb[o];
#pragma unroll
    for (int c = 0; c < 5; ++c) acc = fmaf(lw[o * 5 + c], in5[c], acc);
    h[((size_t)b * HID + o) * PP + p] = acc;
  }
}

// ------------------------------------------------- per-pixel MLP basis gen
__global__ void k_mlp(const float* __restrict__ h,
                      const float* __restrict__ w0, const float* __restrict__ b0,
                      const float* __restrict__ w1, const float* __restrict__ b1,
                      const float* __restrict__ wf, const float* __restrict__ bf,
                      float* __restrict__ Vre, float* __restrict__ Vim) {
  extern __shared__ float smem[];               // wf: 64*512, bf: 512
  float* swf = smem;
  float* sbf = smem + HID * 2 * MM;
  for (int i = threadIdx.x; i < HID * 2 * MM; i += blockDim.x) swf[i] = wf[i];
  for (int i = threadIdx.x; i < 2 * MM; i += blockDim.x) sbf[i] = bf[i];
  __syncthreads();

  int idx = blockIdx.x * blockDim.x + threadIdx.x;
  int b = idx >> 14, p = idx & (PP - 1);
  int ph = p >> 7, pw = p & (WW - 1);

  float in66[66];
  in66[0] = (float)ph * (1.0f / 127.0f);
  in66[1] = (float)pw * (1.0f / 127.0f);
#pragma unroll
  for (int c = 0; c < HID; ++c) in66[2 + c] = h[((size_t)b * HID + c) * PP + p];

  float z1[64];
#pragma unroll
  for (int j = 0; j < 64; ++j) {
    float a = b0[j];
#pragma unroll
    for (int i = 0; i < 66; ++i) a = fmaf(in66[i], w0[i * 64 + j], a);
    z1[j] = gelu_exact(a);
  }
  float z2[64];
#pragma unroll
  for (int j = 0; j < 64; ++j) {
    float a = b1[j];
#pragma unroll
    for (int i = 0; i < 64; ++i) a = fmaf(z1[i], w1[i * 64 + j], a);
    z2[j] = gelu_exact(a);
  }
  for (int m0 = 0; m0 < MM; m0 += 8) {          // rolled chunk loop
    float ar[8], ai[8];
#pragma unroll
    for (int j = 0; j < 8; ++j) { ar[j] = sbf[m0 + j]; ai[j] = sbf[MM + m0 + j]; }
#pragma unroll
    for (int i = 0; i < 64; ++i) {
      float zi = z2[i];
#pragma unroll
      for (int j = 0; j < 8; ++j) {
        ar[j] = fmaf(zi, swf[i * (2 * MM) + m0 + j], ar[j]);
        ai[j] = fmaf(zi, swf[i * (2 * MM) + MM + m0 + j], ai[j]);
      }
    }
#pragma unroll
    for (int j = 0; j < 8; ++j) {
      Vre[((size_t)b * MM + m0 + j) * PP + p] = ar[j];
      Vim[((size_t)b * MM + m0 + j) * PP + p] = ai[j];
    }
  }
}

// ---------------- two-value block reduction: wave32 shuffles + 1 LDS stage
__device__ __forceinline__ void block_reduce2(float& a, float& b) {
  __shared__ float sa[32], sb2[32];
  int lane = threadIdx.x & 31, wid = threadIdx.x >> 5;
#pragma unroll
  for (int off = 16; off > 0; off >>= 1) {
    a += __shfl_xor(a, off, 32);
    b += __shfl_xor(b, off, 32);
  }
  if (lane == 0) { sa[wid] = a; sb2[wid] = b; }
  __syncthreads();
  if (wid == 0) {
    a = sa[lane]; b = sb2[lane];
#pragma unroll
    for (int off = 16; off > 0; off >>= 1) {
      a += __shfl_xor(a, off, 32);
      b += __shfl_xor(b, off, 32);
    }
    if (lane == 0) { sa[0] = a; sb2[0] = b; }
  }
  __syncthreads();
  a = sa[0]; b = sb2[0];
  __syncthreads();
}

// ------------------------------------------------ Gram-Schmidt: init col 0
__global__ void __launch_bounds__(1024) k_gs_init(float* __restrict__ Vre,
                                                  float* __restrict__ Vim) {
  int b = blockIdx.x, t = threadIdx.x;
  v4f* vr = (v4f*)(Vre + (size_t)b * MM * PP);
  v4f* vi = (v4f*)(Vim + (size_t)b * MM * PP);
  v4f re[4], im[4], acc = {};
#pragma unroll
  for (int i = 0; i < 4; ++i) {
    re[i] = vr[t + i * 1024];
    im[i] = vi[t + i * 1024];
    acc += re[i] * re[i] + im[i] * im[i];
  }
  float nn = acc.x + acc.y + acc.z + acc.w, dummy = 0.f;
  block_reduce2(nn, dummy);
  float inv = 1.0f / (sqrtf(nn) + 1e-8f);
#pragma unroll
  for (int i = 0; i < 4; ++i) {
    vr[t + i * 1024] = re[i] * inv;
    vi[t + i * 1024] = im[i] * inv;
  }
}

// -------- Gram-Schmidt step k: update trailing cols, pre-normalize col k+1
__global__ void __launch_bounds__(1024) k_gs_step(float* __restrict__ Vre,
                                                  float* __restrict__ Vim, int k) {
  int b = blockIdx.y, chunk = blockIdx.x, t = threadIdx.x;
  int mStart = chunk * 8; if (mStart < k + 1) mStart = k + 1;
  int mEnd = chunk * 8 + 8; if (mEnd > MM) mEnd = MM;
  if (mStart >= mEnd) return;                       // uniform per block
  const v4f* qr = (const v4f*)(Vre + ((size_t)b * MM + k) * PP);
  const v4f* qi = (const v4f*)(Vim + ((size_t)b * MM + k) * PP);
  v4f qre[4], qim[4];
#pragma unroll
  for (int i = 0; i < 4; ++i) { qre[i] = qr[t + i * 1024]; qim[i] = qi[t + i * 1024]; }
  for (int m = mStart; m < mEnd; ++m) {
    v4f* vr = (v4f*)(Vre + ((size_t)b * MM + m) * PP);
    v4f* vi = (v4f*)(Vim + ((size_t)b * MM + m) * PP);
    v4f re[4], im[4], apr = {}, api = {};
#pragma unroll
    for (int i = 0; i < 4; ++i) {
      re[i] = vr[t + i * 1024];
      im[i] = vi[t + i * 1024];
      apr += qre[i] * re[i] + qim[i] * im[i];       // conj(q) . v
      api += qre[i] * im[i] - qim[i] * re[i];
    }
    float pr = apr.x + apr.y + apr.z + apr.w;
    float pi = api.x + api.y + api.z + api.w;
    block_reduce2(pr, pi);
    v4f as2 = {};
#pragma unroll
    for (int i = 0; i < 4; ++i) {
      re[i] -= pr * qre[i] - pi * qim[i];
      im[i] -= pr * qim[i] + pi * qre[i];
      as2 += re[i] * re[i] + im[i] * im[i];
    }
    float inv = 1.0f;
    if (m == k + 1) {                               // uniform per block
      float nn = as2.x + as2.y + as2.z + as2.w, dummy = 0.f;
      block_reduce2(nn, dummy);
      inv = 1.0f / (sqrtf(nn) + 1e-8f);
    }
#pragma unroll
    for (int i = 0; i < 4; ++i) {
      vr[t + i * 1024] = re[i] * inv;
      vi[t + i * 1024] = im[i] * inv;
    }
  }
}

// ---------------- forward transform: coeff = h . conj(Q)/P  (WMMA, K=16384)
// Double-buffered LDS fed by async global->LDS B128; 3 async ops per thread
// per tile -> s_wait_asynccnt 3 after issuing the next tile.
__global__ void __launch_bounds__(256) k_coeff(const float* __restrict__ h,
    const float* __restrict__ Qre, const float* __restrict__ Qim,
    float* __restrict__ Cr, float* __restrict__ Ci) {
  __shared__ float Hs[2][HID][KCP];
  __shared__ float Qs[2][2][16][KCP];
  int b = blockIdx.y, mt = blockIdx.x;
  int t = threadIdx.x, lane = t & 31, wid = t >> 5;
  int part = wid >> 2, ct = wid & 3;                // wave role: (re/im, c-tile)
  v8f acc = {};
  int arow = ct * 16 + (lane & 15);
  int kloc = (lane >> 4) << 1;
  int mcol = lane & 15;

  // per-thread staging slots (float4 granularity)
  int hrow0 = t >> 3, hrow1 = (t + 256) >> 3, hc4 = (t & 7) << 2;
  int qpp = t >> 7, qrow = (t >> 3) & 15, qc4 = (t & 7) << 2;
  const float* qbase = qpp ? Qim : Qre;

  auto issue = [&](int buf, int kb) {
    ASYNC_COPY_B128(&h[((size_t)b * HID + hrow0) * PP + kb + hc4],
                    &Hs[buf][hrow0][hc4]);
    ASYNC_COPY_B128(&h[((size_t)b * HID + hrow1) * PP + kb + hc4],
                    &Hs[buf][hrow1][hc4]);
    ASYNC_COPY_B128(&qbase[((size_t)b * MM + mt * 16 + qrow) * PP + kb + qc4],
                    &Qs[buf][qpp][qrow][qc4]);
  };

  issue(0, 0);
  for (int kb = 0; kb < PP; kb += KC) {
    int buf = (kb / KC) & 1;
    if (kb + KC < PP) {
      issue(buf ^ 1, kb + KC);                      // overlap next tile
      ASYNC_WAIT(3);                                // current tile landed
    } else {
      ASYNC_WAIT(0);
    }
    __syncthreads();
#pragma unroll
    for (int kk = 0; kk < KC; kk += 4) {
      v2f a, bb;
      a.x = Hs[buf][arow][kk + kloc];
      a.y = Hs[buf][arow][kk + kloc + 1];
      bb.x = Qs[buf][part][mcol][kk + kloc];
      bb.y = Qs[buf][part][mcol][kk + kloc + 1];
      acc = wmma_f32(a, bb, acc);
    }
    __syncthreads();                                // before buffer reuse
  }
  float sc = part ? (-1.0f / (float)PP) : (1.0f / (float)PP);  // conj -> -imag
  float* out = part ? Ci : Cr;
  int mg = mt * 16 + mcol;
  int chi = (lane >> 4) << 3;
#pragma unroll
  for (int r = 0; r < 8; ++r) {
    int c = ct * 16 + r + chi;
    out[((size_t)b * HID + c) * MM + mg] = acc[r] * sc;
  }
}

// ------------------------------------ per-mode channel mixing (tiny, VALU)
__global__ void k_mixed(const float* __restrict__ Cr, const float* __restrict__ Ci,
                        const float* __restrict__ wr, const float* __restrict__ wi,
                        float* __restrict__ Mr, float* __restrict__ Mi) {
  int idx = blockIdx.x * blockDim.x + threadIdx.x;  // over 64*256
  int m = idx & (MM - 1), o = idx >> 8;
  float ar[BB], ai[BB];
#pragma unroll
  for (int b = 0; b < BB; ++b) { ar[b] = 0.f; ai[b] = 0.f; }
  for (int c = 0; c < HID; ++c) {
    float wrv = wr[((size_t)o * HID + c) * MM + m];
    float wiv = wi[((size_t)o * HID + c) * MM + m];
#pragma unroll
    for (int b = 0; b < BB; ++b) {
      float cr = Cr[((size_t)b * HID + c) * MM + m];
      float ci = Ci[((size_t)b * HID + c) * MM + m];
      ar[b] = fmaf(cr, wrv, fmaf(-ci, wiv, ar[b]));
      ai[b] = fmaf(cr, wiv, fmaf(ci, wrv, ai[b]));
    }
  }
#pragma unroll
  for (int b = 0; b < BB; ++b) {
    Mr[((size_t)b * HID + o) * MM + m] = ar[b];
    Mi[((size_t)b * HID + o) * MM + m] = ai[b];
  }
}

// ------- fused: y = Re(M Q) (K=256) + skip (K=64) + bias + LN + GELU (WMMA)
__global__ void __launch_bounds__(256) k_inv(
    const float* __restrict__ Mr, const float* __restrict__ Mi,
    const float* __restrict__ Qre, const float* __restrict__ Qim,
    const float* __restrict__ hin, const float* __restrict__ sw,
    const float* __restrict__ sb, const float* __restrict__ gamma,
    const float* __restrict__ beta, float* __restrict__ hout) {
  int t = threadIdx.x, lane = t & 31, wid = t >> 5;
  int g = blockIdx.x * 8 + wid;                     // 64-pixel group per wave
  int b = g >> 8;
  int pbase = (g & 255) * 64;
  int nidx = lane & 15;
  int kloc = (lane >> 4) << 1;
  int chi = (lane >> 4) << 3;
  v8f acc[4][4] = {};
  // inverse transform: Re((Mr+iMi)(Qr+iQi)) = Mr*Qr - Mi*Qi, K = 256 modes
  for (int kk = 0; kk < MM; kk += 4) {
    int k0 = kk + kloc;
    __builtin_prefetch(&Qre[((size_t)b * MM + k0 + 4) * PP + pbase + nidx], 0, 1);
    __builtin_prefetch(&Qim[((size_t)b * MM + k0 + 4) * PP + pbase + nidx], 0, 1);
    v2f Ar[4], Ai[4];
#pragma unroll
    for (int ctt = 0; ctt < 4; ++ctt) {
      int o = ctt * 16 + nidx;
      const float* mrp = &Mr[((size_t)b * HID + o) * MM + k0];
      const float* mip = &Mi[((size_t)b * HID + o) * MM + k0];
      Ar[ctt].x = mrp[0];  Ar[ctt].y = mrp[1];
      Ai[ctt].x = -mip[0]; Ai[ctt].y = -mip[1];     // fold subtraction into A
    }
#pragma unroll
    for (int pt = 0; pt < 4; ++pt) {
      int pcol = pbase + pt * 16 + nidx;
      v2f Br, Bi;
      Br.x = Qre[((size_t)b * MM + k0) * PP + pcol];
      Br.y = Qre[((size_t)b * MM + k0 + 1) * PP + pcol];
      Bi.x = Qim[((size_t)b * MM + k0) * PP + pcol];
      Bi.y = Qim[((size_t)b * MM + k0 + 1) * PP + pcol];
#pragma unroll
      for (int ctt = 0; ctt < 4; ++ctt) {
        acc[ctt][pt] = wmma_f32(Ar[ctt], Br, acc[ctt][pt]);
        acc[ctt][pt] = wmma_f32(Ai[ctt], Bi, acc[ctt][pt]);
      }
    }
  }
  // 1x1 skip: += sw . hin, K = 64 channels
  for (int kk = 0; kk < HID; kk += 4) {
    int k0 = kk + kloc;
    v2f As[4];
#pragma unroll
    for (int ctt = 0; ctt < 4; ++ctt) {
      int o = ctt * 16 + nidx;
      As[ctt].x = sw[(size_t)o * HID + k0];
      As[ctt].y = sw[(size_t)o * HID + k0 + 1];
    }
#pragma unroll
    for (int pt = 0; pt < 4; ++pt) {
      int pcol = pbase + pt * 16 + nidx;
      v2f Bh;
      Bh.x = hin[((size_t)b * HID + k0) * PP + pcol];
      Bh.y = hin[((size_t)b * HID + k0 + 1) * PP + pcol];
#pragma unroll
      for (int ctt = 0; ctt < 4; ++ctt)
        acc[ctt][pt] = wmma_f32(As[ctt], Bh, acc[ctt][pt]);
    }
  }
  // bias + channel LayerNorm + GELU; a pixel's 64 channels live in lanes l,l^16
#pragma unroll
  for (int pt = 0; pt < 4; ++pt) {
    float s1 = 0.f, s2 = 0.f;
#pragma unroll
    for (int ctt = 0; ctt < 4; ++ctt) {
#pragma unroll
      for (int r = 0; r < 8; ++r) {
        int c = ctt * 16 + r + chi;
        float v = acc[ctt][pt][r] + sb[c];
        acc[ctt][pt][r] = v;
        s1 += v; s2 = fmaf(v, v, s2);
      }
    }
    s1 += __shfl_xor(s1, 16, 32);
    s2 += __shfl_xor(s2, 16, 32);
    float mu = s1 * (1.0f / 64.0f);
    float var = s2 * (1.0f / 64.0f) - mu * mu;
    float rinv = rsqrtf(var + 1e-6f);
    int pcol = pbase + pt * 16 + nidx;
#pragma unroll
    for (int ctt = 0; ctt < 4; ++ctt) {
#pragma unroll
      for (int r = 0; r < 8; ++r) {
        int c = ctt * 16 + r + chi;
        float zn = (acc[ctt][pt][r] - mu) * rinv * gamma[c] + beta[c];
        hout[((size_t)b * HID + c) * PP + pcol] = gelu_exact(zn);
      }
    }
  }
}

// ----------------------------------------------------------- final 64 -> 1
__global__ void k_proj(const float* __restrict__ h, const float* __restrict__ pw,
                       const float* __restrict__ pb, float* __restrict__ out) {
  int idx = blockIdx.x * blockDim.x + threadIdx.x;
  int b = idx >> 14, p = idx & (PP - 1);
  float acc = pb[0];
#pragma unroll 8
  for (int c = 0; c < HID; ++c)
    acc = fmaf(pw[c], h[((size_t)b * HID + c) * PP + p], acc);
  out[idx] = acc;
}

// ===========================================================================
extern "C" void kernel_launch(void* const* d_in, const int* in_sizes, int n_in,
                              void* d_out, int out_size, void* d_ws, size_t ws_size,
                              hipStream_t stream) {
  (void)in_sizes; (void)n_in; (void)out_size; (void)ws_size;
  const float* x      = (const float*)d_in[0];
  const float* lift_w = (const float*)d_in[1];
  const float* lift_b = (const float*)d_in[2];
  const float* w0 = (const float*)d_in[3];
  const float* b0 = (const float*)d_in[4];
  const float* w1 = (const float*)d_in[5];
  const float* b1 = (const float*)d_in[6];
  const float* wf = (const float*)d_in[7];
  const float* bf = (const float*)d_in[8];
  const float* blk_wr[2] = { (const float*)d_in[9],  (const float*)d_in[15] };
  const float* blk_wi[2] = { (const float*)d_in[10], (const float*)d_in[16] };
  const float* blk_sw[2] = { (const float*)d_in[11], (const float*)d_in[17] };
  const float* blk_sb[2] = { (const float*)d_in[12], (const float*)d_in[18] };
  const float* blk_g[2]  = { (const float*)d_in[13], (const float*)d_in[19] };
  const float* blk_b[2]  = { (const float*)d_in[14], (const float*)d_in[20] };
  const float* proj_w = (const float*)d_in[21];
  const float* proj_b = (const float*)d_in[22];
  float* out = (float*)d_out;

  char* ws = (char*)d_ws;
  size_t off = 0;
  auto wsalloc = [&](size_t bytes) -> float* {
    float* p = (float*)(ws + off);
    off += (bytes + 255) & ~(size_t)255;
    return p;
  };
  float* h0  = wsalloc(sizeof(float) * BB * HID * PP);   // 16 MB
  float* hA  = wsalloc(sizeof(float) * BB * HID * PP);   // 16 MB
  float* hB  = wsalloc(sizeof(float) * BB * HID * PP);   // 16 MB
  float* Vre = wsalloc(sizeof(float) * BB * MM * PP);    // 64 MB (L2-resident)
  float* Vim = wsalloc(sizeof(float) * BB * MM * PP);    // 64 MB
  float* Cr  = wsalloc(sizeof(float) * BB * HID * MM);
  float* Ci  = wsalloc(sizeof(float) * BB * HID * MM);
  float* Mrb = wsalloc(sizeof(float) * BB * HID * MM);
  float* Mib = wsalloc(sizeof(float) * BB * HID * MM);

  k_lift<<<dim3(BB * PP / 256), dim3(256), 0, stream>>>(x, lift_w, lift_b, h0);

  size_t shmem = (size_t)(HID * 2 * MM + 2 * MM) * sizeof(float);   // 133 KB
  k_mlp<<<dim3(BB * PP / 256), dim3(256), shmem, stream>>>(
      h0, w0, b0, w1, b1, wf, bf, Vre, Vim);

  k_gs_init<<<dim3(BB), dim3(1024), 0, stream>>>(Vre, Vim);
  for (int k = 0; k + 1 < MM; ++k)
    k_gs_step<<<dim3(MM / 8, BB), dim3(1024), 0, stream>>>(Vre, Vim, k);

  const float* hcur = h0;
  float* houts[2] = { hA, hB };
  for (int blk = 0; blk < 2; ++blk) {
    k_coeff<<<dim3(MM / 16, BB), dim3(256), 0, stream>>>(hcur, Vre, Vim, Cr, Ci);
    k_mixed<<<dim3(HID * MM / 256), dim3(256), 0, stream>>>(
        Cr, Ci, blk_wr[blk], blk_wi[blk], Mrb, Mib);
    k_inv<<<dim3(BB * PP / 512), dim3(256), 0, stream>>>(
        Mrb, Mib, Vre, Vim, hcur, blk_sw[blk], blk_sb[blk], blk_g[blk],
        blk_b[blk], houts[blk]);
    hcur = houts[blk];
  }
  k_proj<<<dim3(BB * PP / 256), dim3(256), 0, stream>>>(hcur, proj_w, proj_b, out);
}